// GAT_17179869190
// MI455X (gfx1250) — compile-verified
//
#include <hip/hip_runtime.h>

typedef __attribute__((ext_vector_type(2))) float v2f;
typedef __attribute__((ext_vector_type(8))) float v8f;

#define IN_DIM   512
#define CH       8      // channels per head, both layers
#define HEADS1   8
#define HEADS2   4
#define L1_OUT   64     // HEADS1*CH
#define L2_OUT   32     // HEADS2*CH
#define NEG_SLOPE 0.2f

// ---------------------------------------------------------------- helpers

__device__ __forceinline__ float atomic_max_float(float* addr, float val) {
  // standard IEEE bit-trick (correct for non-NaN values)
  if (val >= 0.0f)
    return __int_as_float(atomicMax((int*)addr, __float_as_int(val)));
  else
    return __uint_as_float(atomicMin((unsigned int*)addr, __float_as_uint(val)));
}

__device__ __forceinline__ float lrelu(float v) {
  return v >= 0.0f ? v : NEG_SLOPE * v;
}

// ---------------------------------------------------------------- fp32 WMMA GEMM
// C[N, NT*16] = A[N, K] * B[K, NT*16], row-major, exact fp32 via
// V_WMMA_F32_16X16X4_F32. One wave handles a 16-row block and NT column tiles.
template<int K, int NT>
__global__ void gemm_wmma_f32(const float* __restrict__ A,
                              const float* __restrict__ B,
                              float* __restrict__ C, int N) {
  constexpr int NC = NT * 16;
  const int wave = threadIdx.x >> 5;
  const int lane = threadIdx.x & 31;
  const int rowblk = blockIdx.x * (blockDim.x >> 5) + wave;
  if (rowblk * 16 >= N) return;          // wave-uniform: EXEC stays all-ones

  const int half = lane >> 4;            // 0: K=k,k+1   1: K=k+2,k+3
  const int l    = lane & 15;

  int arow = rowblk * 16 + l;
  if (arow >= N) arow = N - 1;           // clamp (only matters in ragged tail)
  const float* __restrict__ Arow = A + (size_t)arow * K;

  v8f acc[NT];
#pragma unroll
  for (int t = 0; t < NT; ++t) acc[t] = (v8f)(0.0f);

  for (int k = 0; k < K; k += 4) {
    const int ka = k + half * 2;
    v2f a;
    a.x = Arow[ka];
    a.y = Arow[ka + 1];
    const float* __restrict__ B0 = B + (size_t)ka * NC;
#pragma unroll
    for (int t = 0; t < NT; ++t) {
      v2f b;
      b.x = B0[t * 16 + l];              // row K=ka
      b.y = B0[NC + t * 16 + l];         // row K=ka+1
      acc[t] = __builtin_amdgcn_wmma_f32_16x16x4_f32(
          /*neg_a=*/false, a, /*neg_b=*/false, b,
          /*c_mod=*/(short)0, acc[t], /*reuse_a=*/false, /*reuse_b=*/false);
    }
  }

  // C/D layout: VGPR r, lanes 0-15 -> M=r, lanes 16-31 -> M=r+8
  const int mbase = rowblk * 16 + half * 8;
  if (rowblk * 16 + 15 < N) {
    // uniform full-tile fast path: no per-lane EXEC manipulation
    float* __restrict__ Crow = C + (size_t)mbase * NC + l;
#pragma unroll
    for (int r = 0; r < 8; ++r) {
#pragma unroll
      for (int t = 0; t < NT; ++t) Crow[t * 16] = acc[t][r];
      Crow += NC;
    }
  } else {
#pragma unroll
    for (int r = 0; r < 8; ++r) {
      const int m = mbase + r;
      if (m < N) {
        float* __restrict__ Crow = C + (size_t)m * NC;
#pragma unroll
        for (int t = 0; t < NT; ++t) Crow[t * 16 + l] = acc[t][r];
      }
    }
  }
}

// ---------------------------------------------------------------- attention prep
// per (node, head): e_src/e_dst dots, init max=-inf, denom=0, zero agg slice.
__global__ void prep_kernel(const float* __restrict__ h,
                            const float* __restrict__ a_src,
                            const float* __restrict__ a_dst,
                            float* __restrict__ es, float* __restrict__ ed,
                            float* __restrict__ m,  float* __restrict__ den,
                            float* __restrict__ agg, int N, int heads) {
  const int idx = blockIdx.x * blockDim.x + threadIdx.x;   // n*heads + hh
  if (idx >= N * heads) return;
  const int hh = idx % heads;
  const float4* __restrict__ hp4 = (const float4*)(h + (size_t)idx * CH);
  const float4* __restrict__ as4 = (const float4*)(a_src + hh * CH);
  const float4* __restrict__ ad4 = (const float4*)(a_dst + hh * CH);
  const float4 h0 = hp4[0], h1 = hp4[1];
  const float4 s0 = as4[0], s1 = as4[1];
  const float4 d0 = ad4[0], d1 = ad4[1];
  float s = h0.x * s0.x + h0.y * s0.y + h0.z * s0.z + h0.w * s0.w
          + h1.x * s1.x + h1.y * s1.y + h1.z * s1.z + h1.w * s1.w;
  float d = h0.x * d0.x + h0.y * d0.y + h0.z * d0.z + h0.w * d0.w
          + h1.x * d1.x + h1.y * d1.y + h1.z * d1.z + h1.w * d1.w;
  es[idx] = s;
  ed[idx] = d;
  m[idx]  = -1e30f;
  den[idx] = 0.0f;
  float4* __restrict__ ag4 = (float4*)(agg + (size_t)idx * CH);
  ag4[0] = make_float4(0.f, 0.f, 0.f, 0.f);
  ag4[1] = make_float4(0.f, 0.f, 0.f, 0.f);
}

// ---------------------------------------------------------------- edge pass 1: segment max
template<int HEADS>
__global__ void edge_max_kernel(const int* __restrict__ src,
                                const int* __restrict__ dst,
                                int E, int N,
                                const float* __restrict__ es,
                                const float* __restrict__ ed,
                                float* __restrict__ m) {
  const int e = blockIdx.x * blockDim.x + threadIdx.x;
  if (e >= E + N) return;
  int s, d;
  if (e < E) { s = src[e]; d = dst[e]; }
  else       { s = e - E;  d = s; }     // self loop
  const float4* __restrict__ es4 = (const float4*)(es + (size_t)s * HEADS);
  const float4* __restrict__ ed4 = (const float4*)(ed + (size_t)d * HEADS);
  float* __restrict__ mp = m + (size_t)d * HEADS;
#pragma unroll
  for (int q = 0; q < HEADS / 4; ++q) {
    const float4 a = es4[q];
    const float4 b = ed4[q];
    atomic_max_float(&mp[q * 4 + 0], lrelu(a.x + b.x));
    atomic_max_float(&mp[q * 4 + 1], lrelu(a.y + b.y));
    atomic_max_float(&mp[q * 4 + 2], lrelu(a.z + b.z));
    atomic_max_float(&mp[q * 4 + 3], lrelu(a.w + b.w));
  }
}

// ---------------------------------------------------------------- edge pass 2: exp-sum + weighted scatter
template<int HEADS>
__global__ void edge_acc_kernel(const int* __restrict__ src,
                                const int* __restrict__ dst,
                                int E, int N,
                                const float* __restrict__ es,
                                const float* __restrict__ ed,
                                const float* __restrict__ m,
                                const float* __restrict__ h,
                                float* __restrict__ den,
                                float* __restrict__ agg) {
  const int e = blockIdx.x * blockDim.x + threadIdx.x;
  if (e >= E + N) return;
  int s, d;
  if (e < E) { s = src[e]; d = dst[e]; }
  else       { s = e - E;  d = s; }
  const float4* __restrict__ es4 = (const float4*)(es + (size_t)s * HEADS);
  const float4* __restrict__ ed4 = (const float4*)(ed + (size_t)d * HEADS);
  const float4* __restrict__ m4  = (const float4*)(m  + (size_t)d * HEADS);
  float* __restrict__ dp = den + (size_t)d * HEADS;
#pragma unroll
  for (int q = 0; q < HEADS / 4; ++q) {
    const float4 a  = es4[q];
    const float4 b  = ed4[q];
    const float4 mm = m4[q];
    float ex[4];
    ex[0] = expf(lrelu(a.x + b.x) - mm.x);
    ex[1] = expf(lrelu(a.y + b.y) - mm.y);
    ex[2] = expf(lrelu(a.z + b.z) - mm.z);
    ex[3] = expf(lrelu(a.w + b.w) - mm.w);
#pragma unroll
    for (int j = 0; j < 4; ++j) {
      const int hh = q * 4 + j;
      atomicAdd(&dp[hh], ex[j]);
      const float4* __restrict__ hs4 =
          (const float4*)(h + ((size_t)s * HEADS + hh) * CH);
      const float4 h0 = hs4[0], h1 = hs4[1];
      float* __restrict__ ag = agg + ((size_t)d * HEADS + hh) * CH;
      atomicAdd(&ag[0], h0.x * ex[j]);
      atomicAdd(&ag[1], h0.y * ex[j]);
      atomicAdd(&ag[2], h0.z * ex[j]);
      atomicAdd(&ag[3], h0.w * ex[j]);
      atomicAdd(&ag[4], h1.x * ex[j]);
      atomicAdd(&ag[5], h1.y * ex[j]);
      atomicAdd(&ag[6], h1.z * ex[j]);
      atomicAdd(&ag[7], h1.w * ex[j]);
    }
  }
}

// ---------------------------------------------------------------- layer-1 finalize: /denom + bias, ELU (in place)
__global__ void fin1_kernel(float* __restrict__ agg,
                            const float* __restrict__ den,
                            const float* __restrict__ b, int N) {
  const int idx = blockIdx.x * blockDim.x + threadIdx.x;   // n*64 + j
  if (idx >= N * L1_OUT) return;
  const int j  = idx & (L1_OUT - 1);
  const int nh = idx >> 3;                                 // n*HEADS1 + head
  float v = agg[idx] / (den[nh] + 1e-16f) + b[j];
  agg[idx] = v > 0.0f ? v : (expf(v) - 1.0f);              // ELU
}

// ---------------------------------------------------------------- layer-2 finalize: /denom, mean over heads, + bias
__global__ void fin2_kernel(const float* __restrict__ agg,
                            const float* __restrict__ den,
                            const float* __restrict__ b,
                            float* __restrict__ out, int N) {
  const int idx = blockIdx.x * blockDim.x + threadIdx.x;   // n*8 + dch
  if (idx >= N * CH) return;
  const int n   = idx >> 3;
  const int dch = idx & 7;
  float acc = 0.0f;
#pragma unroll
  for (int hh = 0; hh < HEADS2; ++hh)
    acc += agg[(size_t)n * L2_OUT + hh * CH + dch] / (den[n * HEADS2 + hh] + 1e-16f);
  out[idx] = acc * (1.0f / HEADS2) + b[dch];
}

// ---------------------------------------------------------------- launch

extern "C" void kernel_launch(void* const* d_in, const int* in_sizes, int n_in,
                              void* d_out, int out_size, void* d_ws, size_t ws_size,
                              hipStream_t stream) {
  const float* x   = (const float*)d_in[0];
  const int*   ei  = (const int*)d_in[1];
  const float* W1  = (const float*)d_in[2];
  const float* as1 = (const float*)d_in[3];
  const float* ad1 = (const float*)d_in[4];
  const float* b1  = (const float*)d_in[5];
  const float* W2  = (const float*)d_in[6];
  const float* as2 = (const float*)d_in[7];
  const float* ad2 = (const float*)d_in[8];
  const float* b2  = (const float*)d_in[9];
  float* out = (float*)d_out;

  const int N = in_sizes[0] / IN_DIM;
  const int E = in_sizes[1] / 2;
  const int* src = ei;
  const int* dst = ei + E;
  const int ET = E + N;

  // workspace carve-up (floats); every offset is a multiple of N*4 floats
  // (N*16 bytes), so all sub-buffers are 16B-aligned for float4 access.
  float* p = (float*)d_ws;
  float* h1   = p; p += (size_t)N * L1_OUT;
  float* es1  = p; p += (size_t)N * HEADS1;
  float* ed1  = p; p += (size_t)N * HEADS1;
  float* m1   = p; p += (size_t)N * HEADS1;
  float* den1 = p; p += (size_t)N * HEADS1;
  float* agg1 = p; p += (size_t)N * L1_OUT;   // becomes x2 after fin1 (in place)
  float* h2   = p; p += (size_t)N * L2_OUT;
  float* es2  = p; p += (size_t)N * HEADS2;
  float* ed2  = p; p += (size_t)N * HEADS2;
  float* m2   = p; p += (size_t)N * HEADS2;
  float* den2 = p; p += (size_t)N * HEADS2;
  float* agg2 = p; p += (size_t)N * L2_OUT;

  const int rowblks = (N + 15) / 16;
  const dim3 gblk(256);                       // 8 waves per block
  const dim3 ggrid((rowblks + 7) / 8);
  const int T = 256;

  // ---- layer 1
  gemm_wmma_f32<IN_DIM, 4><<<ggrid, gblk, 0, stream>>>(x, W1, h1, N);
  prep_kernel<<<(N * HEADS1 + T - 1) / T, T, 0, stream>>>(
      h1, as1, ad1, es1, ed1, m1, den1, agg1, N, HEADS1);
  edge_max_kernel<HEADS1><<<(ET + T - 1) / T, T, 0, stream>>>(
      src, dst, E, N, es1, ed1, m1);
  edge_acc_kernel<HEADS1><<<(ET + T - 1) / T, T, 0, stream>>>(
      src, dst, E, N, es1, ed1, m1, h1, den1, agg1);
  fin1_kernel<<<(N * L1_OUT + T - 1) / T, T, 0, stream>>>(agg1, den1, b1, N);

  // ---- layer 2 (input = agg1, now ELU'd features)
  gemm_wmma_f32<L1_OUT, 2><<<ggrid, gblk, 0, stream>>>(agg1, W2, h2, N);
  prep_kernel<<<(N * HEADS2 + T - 1) / T, T, 0, stream>>>(
      h2, as2, ad2, es2, ed2, m2, den2, agg2, N, HEADS2);
  edge_max_kernel<HEADS2><<<(ET + T - 1) / T, T, 0, stream>>>(
      src, dst, E, N, es2, ed2, m2);
  edge_acc_kernel<HEADS2><<<(ET + T - 1) / T, T, 0, stream>>>(
      src, dst, E, N, es2, ed2, m2, h2, den2, agg2);
  fin2_kernel<<<(N * CH + T - 1) / T, T, 0, stream>>>(agg2, den2, b2, out, N);
}